// MultiBoxLoss_12352325944174
// MI455X (gfx1250) — compile-verified
//
#include <hip/hip_runtime.h>

typedef __attribute__((ext_vector_type(2))) float v2f;
typedef __attribute__((ext_vector_type(8))) float v8f;

#define NB 32
#define NA 32768
#define NG 32
#define NBLK 128 /* NA / 256 */

__device__ __forceinline__ float smoothl1(float d) {
  float ad = fabsf(d);
  return ad < 1.0f ? 0.5f * d * d : ad - 0.5f;
}

__device__ __forceinline__ float warp_sum(float v) {
  v += __shfl_xor(v, 16, 32);
  v += __shfl_xor(v, 8, 32);
  v += __shfl_xor(v, 4, 32);
  v += __shfl_xor(v, 2, 32);
  v += __shfl_xor(v, 1, 32);
  return v;
}

// Async copy of 16 bytes global -> LDS (CDNA5 GLOBAL_LOAD_ASYNC_TO_LDS_B128,
// tracked by ASYNCcnt). lds_off is the workgroup-relative LDS byte offset
// (low 32 bits of the generic pointer); gaddr is the 64-bit global address.
__device__ __forceinline__ void async_copy_b128(unsigned lds_off,
                                                const void* gaddr) {
  asm volatile("global_load_async_to_lds_b128 %0, %1, off"
               :
               : "v"(lds_off), "v"((unsigned long long)(size_t)gaddr)
               : "memory");
}
__device__ __forceinline__ void wait_asynccnt0() {
  asm volatile("s_wait_asynccnt 0" ::: "memory");
}

// Kernel 1: per-anchor matching, loc loss, CE terms, per-block partial sums.
// grid = (NBLK, NB), block = 256 (8 wave32 waves).
__global__ __launch_bounds__(256) void mbox_match_kernel(
    const float4* __restrict__ loc, const float2* __restrict__ conf,
    const float4* __restrict__ anchors, const float4* __restrict__ gt,
    unsigned* __restrict__ neg_key, float* __restrict__ neg_ce,
    float* __restrict__ partials) {
  __shared__ float4 s_gt[NG];
  __shared__ float s_areag[NG];
  __shared__ float s_red[32];
  const int tid = threadIdx.x;
  const int b = blockIdx.y;
  const int a = blockIdx.x * 256 + tid;

  // Stage gt boxes via the CDNA5 async-to-LDS path (wave 0, 32 lanes x 16B).
  if (tid < NG) {
    unsigned lds_off = (unsigned)(size_t)(&s_gt[tid]);
    async_copy_b128(lds_off, gt + b * NG + tid);
    wait_asynccnt0();
    float4 g = s_gt[tid];
    s_areag[tid] = (g.z - g.x) * (g.w - g.y);
  }
  __syncthreads();

  float4 an = anchors[a];
  float ax0 = an.x - 0.5f * an.z, ay0 = an.y - 0.5f * an.w;
  float ax1 = an.x + 0.5f * an.z, ay1 = an.y + 0.5f * an.w;
  float area_a = an.z * an.w;

  float best = -1.0f;
  int bi = 0;
#pragma unroll
  for (int g = 0; g < NG; ++g) {
    float4 gb = s_gt[g];
    float lx = fmaxf(ax0, gb.x), ly = fmaxf(ay0, gb.y);
    float rx = fminf(ax1, gb.z), ry = fminf(ay1, gb.w);
    float w = fmaxf(rx - lx, 0.0f), h = fmaxf(ry - ly, 0.0f);
    float inter = w * h;
    float iou = inter / (area_a + s_areag[g] - inter);
    if (iou > best) { best = iou; bi = g; }  // first-max == jnp.argmax
  }

  bool pos = best >= 0.35f;
  bool neg = best < 0.10f;
  size_t ia = (size_t)b * NA + a;
  float2 cf = conf[ia];
  float mx = fmaxf(cf.x, cf.y);
  float lz = mx + logf(expf(cf.x - mx) + expf(cf.y - mx));

  float sl1 = 0.0f;
  if (pos) {
    float4 mb = s_gt[bi];
    float4 lc = loc[ia];
    float ex = (0.5f * (mb.x + mb.z) - an.x) / an.z;
    float ey = (0.5f * (mb.y + mb.w) - an.y) / an.w;
    float ew = logf(fmaxf(mb.z - mb.x, 1e-8f) / an.z);
    float eh = logf(fmaxf(mb.w - mb.y, 1e-8f) / an.w);
    sl1 = smoothl1(lc.x - ex) + smoothl1(lc.y - ey) +
          smoothl1(lc.z - ew) + smoothl1(lc.w - eh);
  }
  float cepos = pos ? (lz - cf.y) : 0.0f;

  // Order-preserving float->uint key for radix select; sentinel for non-neg.
  unsigned key = 0xFFFFFFFFu;
  if (neg) {
    unsigned bb = __float_as_uint(cf.x);
    key = bb ^ ((unsigned)((int)bb >> 31) | 0x80000000u);
  }
  neg_key[ia] = key;
  neg_ce[ia] = lz - cf.x;

  // Block reduce 4 quantities: sl1, ce_pos, n_pos, n_neg.
  float q0 = warp_sum(sl1);
  float q1 = warp_sum(cepos);
  float q2 = warp_sum(pos ? 1.0f : 0.0f);
  float q3 = warp_sum(neg ? 1.0f : 0.0f);
  int lane = tid & 31, w = tid >> 5;
  if (lane == 0) {
    s_red[w * 4 + 0] = q0; s_red[w * 4 + 1] = q1;
    s_red[w * 4 + 2] = q2; s_red[w * 4 + 3] = q3;
  }
  __syncthreads();

  // Cross-wave combine via WMMA f32 16x16x4: A = ones, B VGPR0 = partials,
  // B VGPR1 = 0  =>  D[m][n] = x[n] + x[n+16]  (full f32 precision).
  if (tid < 32) {
    v2f bmat; bmat.x = s_red[tid]; bmat.y = 0.0f;
    v2f amat; amat.x = 1.0f; amat.y = 1.0f;
    v8f cacc = {};
    v8f d = __builtin_amdgcn_wmma_f32_16x16x4_f32(
        false, amat, false, bmat, (short)0, cacc, false, false);
    float y = d[0];
    y += __shfl_xor(y, 4, 32);
    y += __shfl_xor(y, 8, 32);
    if (tid < 4)
      partials[((size_t)b * NBLK + blockIdx.x) * 4 + tid] = y;
  }
}

// Kernel 2: per-batch radix select (k-th smallest bg key) + selected-neg CE
// sum. grid = NB blocks, 256 threads. The full 32768-key row (128 KB) is
// staged once into LDS via async b128 copies; all 5 passes then run on LDS.
__global__ __launch_bounds__(256) void mbox_select_kernel(
    const unsigned* __restrict__ neg_key, const float* __restrict__ neg_ce,
    const float* __restrict__ partials, float* __restrict__ batch_out) {
  extern __shared__ unsigned s_keys[];  // NA keys = 128 KB dynamic LDS
  __shared__ unsigned s_hist[256];
  __shared__ float s_tot[4];
  __shared__ unsigned s_prefix;
  __shared__ int s_want;
  __shared__ unsigned s_taken;
  __shared__ float s_red[256];
  const int tid = threadIdx.x;
  const int b = blockIdx.x;

  // Async-stage keys: 256 lanes x 16B x 32 iterations = 128 KB.
  {
    const unsigned* gk = neg_key + (size_t)b * NA;
    for (int base = tid * 4; base < NA; base += 256 * 4) {
      unsigned lds_off = (unsigned)(size_t)(&s_keys[base]);
      async_copy_b128(lds_off, gk + base);
    }
    wait_asynccnt0();
  }

  if (tid < 4) {
    float s = 0.0f;
    for (int k = 0; k < NBLK; ++k)  // deterministic fixed-order sum
      s += partials[((size_t)b * NBLK + k) * 4 + tid];
    s_tot[tid] = s;
  }
  __syncthreads();  // keys staged by all waves + totals visible

  int npos = (int)(s_tot[2] + 0.5f);
  int nneg = (int)(s_tot[3] + 0.5f);
  int k3 = 3 * npos;
  int ksel = k3 < nneg ? k3 : nneg;

  const float* ces = neg_ce + (size_t)b * NA;
  float ce_sum = 0.0f;

  if (ksel > 0) {
    if (tid == 0) { s_prefix = 0u; s_want = ksel; }
    __syncthreads();
    for (int shift = 24; shift >= 0; shift -= 8) {
      s_hist[tid] = 0u;
      __syncthreads();
      unsigned prefix = s_prefix;
      for (int i = tid; i < NA; i += 256) {
        unsigned key = s_keys[i];
        bool match = (shift == 24) ||
                     ((key >> (shift + 8)) == (prefix >> (shift + 8)));
        if (match) atomicAdd(&s_hist[(key >> shift) & 0xFFu], 1u);
      }
      __syncthreads();
      if (tid == 0) {
        int want = s_want;
        unsigned cum = 0;
        int dsel = 255;
        for (int dg = 0; dg < 256; ++dg) {
          unsigned c = s_hist[dg];
          if (cum + c >= (unsigned)want) { dsel = dg; break; }
          cum += c;
        }
        s_prefix = prefix | ((unsigned)dsel << shift);
        s_want = want - (int)cum;
      }
      __syncthreads();
    }
    unsigned T = s_prefix;
    int extra = s_want;  // how many key==T elements to take
    if (tid == 0) s_taken = 0u;
    __syncthreads();
    for (int i = tid; i < NA; i += 256) {
      unsigned key = s_keys[i];
      if (key < T) {
        ce_sum += ces[i];
      } else if (key == T) {
        unsigned old = atomicAdd(&s_taken, 1u);
        if (old < (unsigned)extra) ce_sum += ces[i];
      }
    }
  }

  s_red[tid] = ce_sum;
  __syncthreads();
  for (int s = 128; s > 0; s >>= 1) {
    if (tid < s) s_red[tid] += s_red[tid + s];
    __syncthreads();
  }
  if (tid == 0) {
    batch_out[0 * NB + b] = s_tot[0];     // sum smooth-L1 (pos)
    batch_out[1 * NB + b] = s_tot[1];     // sum CE pos
    batch_out[2 * NB + b] = s_red[0];     // sum CE selected neg
    batch_out[3 * NB + b] = s_tot[2];     // pos count
    batch_out[4 * NB + b] = (float)ksel;  // selected-neg count
  }
}

// Kernel 3: combine 32 batches -> two scalar losses. One wave32.
__global__ void mbox_final_kernel(const float* __restrict__ batch_out,
                                  float* __restrict__ out) {
  int tid = threadIdx.x;  // 32 threads, one per batch
  float sl1 = warp_sum(batch_out[0 * NB + tid]);
  float cep = warp_sum(batch_out[1 * NB + tid]);
  float cen = warp_sum(batch_out[2 * NB + tid]);
  float pc = warp_sum(batch_out[3 * NB + tid]);
  float nc = warp_sum(batch_out[4 * NB + tid]);
  if (tid == 0) {
    out[0] = sl1 / fmaxf(pc * 4.0f, 1.0f);
    out[1] = (cep + cen) / fmaxf(pc + nc, 1.0f);
  }
}

extern "C" void kernel_launch(void* const* d_in, const int* in_sizes, int n_in,
                              void* d_out, int out_size, void* d_ws,
                              size_t ws_size, hipStream_t stream) {
  const float4* loc = (const float4*)d_in[0];      // (B,A,4) f32
  const float2* conf = (const float2*)d_in[1];     // (B,A,2) f32
  const float4* anchors = (const float4*)d_in[2];  // (A,4) f32
  const float4* gt = (const float4*)d_in[3];       // (B,G,4) f32

  char* ws = (char*)d_ws;
  unsigned* neg_key = (unsigned*)ws;                          // NB*NA*4 = 4 MB
  float* neg_ce = (float*)(ws + (size_t)NB * NA * 4);         // 4 MB
  float* partials = (float*)(ws + (size_t)NB * NA * 8);       // NB*NBLK*4 f32
  float* batch_out =
      (float*)(ws + (size_t)NB * NA * 8 + (size_t)NB * NBLK * 4 * 4);

  dim3 g1(NBLK, NB);
  mbox_match_kernel<<<g1, 256, 0, stream>>>(loc, conf, anchors, gt, neg_key,
                                            neg_ce, partials);
  mbox_select_kernel<<<NB, 256, NA * sizeof(unsigned), stream>>>(
      neg_key, neg_ce, partials, batch_out);
  mbox_final_kernel<<<1, 32, 0, stream>>>(batch_out, (float*)d_out);
}